// Sinkhorn_4569845203050
// MI455X (gfx1250) — compile-verified
//
#include <hip/hip_runtime.h>
#include <hip/hip_bf16.h>

// Debiased Sinkhorn divergence, N=M=8192, D=16, eps=0.0025, 50 symmetric iters.
//
// Per iteration: 4 softmin reductions over 8192x8192 logits. Cost dots are
// recomputed on the fly with chained V_WMMA_F32_16X16X4_F32 (fp32 WMMA: the
// 1/eps=400 logit scaling forbids f16/bf16 inputs). x/y (512KB each) stay in
// L2 (192MB), so the kernel is transcendental-bound: 1 v_exp_f32 per logit
// (13.4G exp2 total) via a branchless base-2 online LSE.

typedef __attribute__((ext_vector_type(2))) float v2f;
typedef __attribute__((ext_vector_type(8))) float v8f;

#define NPT 8192
#define DIM 16

// workspace layout (floats)
constexpr int HX_OFF  = 0;              // 0.5*|x_i|^2            [NPT]
constexpr int HY_OFF  = NPT;            // 0.5*|y_j|^2            [NPT]
constexpr int POT_OFF = 2 * NPT;        // f0,f1,g0,g1,sx0,sx1,sy0,sy1 [8*NPT]
constexpr int U_OFF   = POT_OFF + 8 * NPT; // u vectors per problem    [4*NPT]
// total = 14*NPT floats = 448 KB

constexpr float EPS     = 0.0025f;                 // blur^2
constexpr float LOG2E   = 1.4426950408889634f;
constexpr float SCALE2  = LOG2E / EPS;             // log2e/eps = 577.078...
constexpr float UCONST  = -13.0f;                  // log2e * log(1/8192) = -log2(8192)
constexpr float EPSLN2  = EPS * 0.6931471805599453f;

__device__ __forceinline__ float neg_inf() { return -__builtin_inff(); }

// ---------------------------------------------------------------- init
__global__ __launch_bounds__(256) void sinkhorn_init(
    const float* __restrict__ x, const float* __restrict__ y,
    float* __restrict__ ws) {
  int i = blockIdx.x * 256 + threadIdx.x;
  float sx = 0.f, sy = 0.f;
#pragma unroll
  for (int k = 0; k < DIM; ++k) {
    float xv = x[i * DIM + k];
    float yv = y[i * DIM + k];
    sx = fmaf(xv, xv, sx);
    sy = fmaf(yv, yv, sy);
  }
  ws[HX_OFF + i] = 0.5f * sx;
  ws[HY_OFF + i] = 0.5f * sy;
  // zero pot buffer 0 (f0,g0,sx0,sy0 = slots 0,2,4,6)
  ws[POT_OFF + 0 * NPT + i] = 0.f;
  ws[POT_OFF + 2 * NPT + i] = 0.f;
  ws[POT_OFF + 4 * NPT + i] = 0.f;
  ws[POT_OFF + 6 * NPT + i] = 0.f;
}

// ------------------------------------------------------- per-iter u vectors
// u_p[j] = log2e*(log w) + (pot[j] - h[j]) * log2e/eps  (base-2 logit bias)
__global__ __launch_bounds__(256) void sinkhorn_prep(
    float* __restrict__ ws, int cur) {
  int j = blockIdx.x * 256 + threadIdx.x;
  float hx = ws[HX_OFF + j], hy = ws[HY_OFF + j];
  float f  = ws[POT_OFF + (0 + cur) * NPT + j];
  float g  = ws[POT_OFF + (2 + cur) * NPT + j];
  float sx = ws[POT_OFF + (4 + cur) * NPT + j];
  float sy = ws[POT_OFF + (6 + cur) * NPT + j];
  ws[U_OFF + 0 * NPT + j] = fmaf(g - hy, SCALE2, UCONST);  // f-update (cols y)
  ws[U_OFF + 1 * NPT + j] = fmaf(f - hx, SCALE2, UCONST);  // g-update (cols x)
  ws[U_OFF + 2 * NPT + j] = fmaf(sx - hx, SCALE2, UCONST); // sx-update
  ws[U_OFF + 3 * NPT + j] = fmaf(sy - hy, SCALE2, UCONST); // sy-update
}

// ------------------------------------------------------------- main softmin
// grid = (NPT/16 row-tiles, 4 problems), block = 128 (4 waves).
// Each wave: resident A tile (16 rows x 16 dims), streams 16-col chunks,
// 4x V_WMMA_F32_16X16X4_F32 per chunk, online base-2 LSE per C/D VGPR row.
__global__ __launch_bounds__(128) void sinkhorn_softmin(
    const float* __restrict__ xg, const float* __restrict__ yg,
    float* __restrict__ ws, int cur) {
  const int p    = blockIdx.y;
  const int tid  = threadIdx.x;
  const int lane = tid & 31;
  const int w    = tid >> 5;
  const int rsel = lane & 15;          // row (A) / col (B) select within tile
  const int kb   = (lane >> 4) << 1;   // K sub-offset: 0 for lanes<16, 2 else

  const float* rowsrc = (p == 1 || p == 3) ? yg : xg;
  const float* colsrc = (p == 1 || p == 2) ? xg : yg;
  const float* hrow   = ws + ((p == 1 || p == 3) ? HY_OFF : HX_OFF);
  const float* uvec   = ws + U_OFF + p * NPT;
  const float* pin    = ws + POT_OFF + (2 * p + cur) * NPT;
  float*       pout   = ws + POT_OFF + (2 * p + (cur ^ 1)) * NPT;

  const int rowbase = blockIdx.x * 16;

  // A-matrix (16x4 f32 per wmma, 4 chained for K=16), ISA VGPR layout:
  // lanes 0-15 hold K = k0,k0+1 ; lanes 16-31 hold K = k0+2,k0+3.
  const float* arow = rowsrc + (rowbase + rsel) * DIM + kb;
  v2f a0 = *(const v2f*)(arow + 0);
  v2f a1 = *(const v2f*)(arow + 4);
  v2f a2 = *(const v2f*)(arow + 8);
  v2f a3 = *(const v2f*)(arow + 12);

  float m[8], s[8];
#pragma unroll
  for (int v = 0; v < 8; ++v) { m[v] = neg_inf(); s[v] = 0.f; }

  for (int c = w; c < NPT / 16; c += 4) {
    const int jb = c * 16;
    const float* brow = colsrc + (jb + rsel) * DIM + kb;  // same layout as A
    v2f b0 = *(const v2f*)(brow + 0);
    v2f b1 = *(const v2f*)(brow + 4);
    v2f b2 = *(const v2f*)(brow + 8);
    v2f b3 = *(const v2f*)(brow + 12);
    float uv = uvec[jb + rsel];

    v8f acc = {0.f, 0.f, 0.f, 0.f, 0.f, 0.f, 0.f, 0.f};
    acc = __builtin_amdgcn_wmma_f32_16x16x4_f32(false, a0, false, b0, (short)0, acc, false, false);
    acc = __builtin_amdgcn_wmma_f32_16x16x4_f32(false, a1, false, b1, (short)0, acc, false, false);
    acc = __builtin_amdgcn_wmma_f32_16x16x4_f32(false, a2, false, b2, (short)0, acc, false, false);
    acc = __builtin_amdgcn_wmma_f32_16x16x4_f32(false, a3, false, b3, (short)0, acc, false, false);

    // C/D layout: VGPR v -> row (rowbase+v) for lanes<16, (rowbase+8+v) else;
    // col = jb + (lane&15) for both halves -> uv is the right bias everywhere.
#pragma unroll
    for (int v = 0; v < 8; ++v) {
      float arg = fmaf(acc[v], SCALE2, uv);     // base-2 logit
      float d   = m[v] - arg;
      float e   = __builtin_amdgcn_exp2f(-__builtin_fabsf(d));
      float s_keep = s[v] + e;                  // m unchanged: add exp2(arg-m)
      float s_new  = fmaf(s[v], e, 1.0f);       // new max:     rescale + 1
      s[v] = (d >= 0.f) ? s_keep : s_new;
      m[v] = fmaxf(m[v], arg);
    }
  }

  // merge (m,s) across the 16 lanes of each half (cols are strided per lane)
#pragma unroll
  for (int v = 0; v < 8; ++v) {
#pragma unroll
    for (int off = 1; off < 16; off <<= 1) {
      float mo = __shfl_xor(m[v], off, 32);
      float so = __shfl_xor(s[v], off, 32);
      float nm = fmaxf(m[v], mo);
      s[v] = s[v] * __builtin_amdgcn_exp2f(m[v] - nm) +
             so   * __builtin_amdgcn_exp2f(mo - nm);
      m[v] = nm;
    }
  }

  __shared__ float lm[4][16];
  __shared__ float lsum[4][16];
  if (lane == 0) {
#pragma unroll
    for (int v = 0; v < 8; ++v) { lm[w][v] = m[v]; lsum[w][v] = s[v]; }
  }
  if (lane == 16) {
#pragma unroll
    for (int v = 0; v < 8; ++v) { lm[w][8 + v] = m[v]; lsum[w][8 + v] = s[v]; }
  }
  __syncthreads();

  if (tid < 16) {
    float M = neg_inf(), S = 0.f;
#pragma unroll
    for (int ww = 0; ww < 4; ++ww) {
      float mo = lm[ww][tid], so = lsum[ww][tid];
      float nm = fmaxf(M, mo);
      S = S * __builtin_amdgcn_exp2f(M - nm) +
          so * __builtin_amdgcn_exp2f(mo - nm);
      M = nm;
    }
    float lse2 = M + __builtin_amdgcn_logf(S);  // v_log_f32 = log2
    int r = rowbase + tid;
    // softmin = h_i - eps*ln2*lse2 ; averaged update: 0.5*(old + softmin)
    pout[r] = 0.5f * (pin[r] + hrow[r] - EPSLN2 * lse2);
  }
}

// ---------------------------------------------------------------- reduce
__global__ __launch_bounds__(256) void sinkhorn_final(
    const float* __restrict__ ws, float* __restrict__ out, int fin) {
  __shared__ float red[256];
  float acc = 0.f;
  for (int i = threadIdx.x; i < NPT; i += 256) {
    acc += ws[POT_OFF + (0 + fin) * NPT + i]   // f
         + ws[POT_OFF + (2 + fin) * NPT + i]   // g
         - ws[POT_OFF + (4 + fin) * NPT + i]   // sx
         - ws[POT_OFF + (6 + fin) * NPT + i];  // sy
  }
  red[threadIdx.x] = acc;
  __syncthreads();
  for (int st = 128; st > 0; st >>= 1) {
    if (threadIdx.x < st) red[threadIdx.x] += red[threadIdx.x + st];
    __syncthreads();
  }
  if (threadIdx.x == 0) out[0] = red[0] * (1.0f / (float)NPT);
}

extern "C" void kernel_launch(void* const* d_in, const int* in_sizes, int n_in,
                              void* d_out, int out_size, void* d_ws, size_t ws_size,
                              hipStream_t stream) {
  const float* x = (const float*)d_in[0];
  const float* y = (const float*)d_in[1];
  float* ws = (float*)d_ws;   // needs 14*8192*4 = 448 KB
  float* out = (float*)d_out;

  sinkhorn_init<<<NPT / 256, 256, 0, stream>>>(x, y, ws);
  for (int it = 0; it < 50; ++it) {
    int cur = it & 1;
    sinkhorn_prep<<<NPT / 256, 256, 0, stream>>>(ws, cur);
    dim3 grid(NPT / 16, 4);
    sinkhorn_softmin<<<grid, 128, 0, stream>>>(x, y, ws, cur);
  }
  sinkhorn_final<<<1, 256, 0, stream>>>(ws, out, /*fin=*/0); // 50 iters -> buf 0
}